// EdgeUpdate_69999376990299
// MI455X (gfx1250) — compile-verified
//
#include <hip/hip_runtime.h>
#include <hip/hip_bf16.h>

// ---------------------------------------------------------------------------
// Problem constants (from the reference)
// ---------------------------------------------------------------------------
#define B_    4
#define NAT_  64
#define NBR_  24
#define FN_   128
#define FE_   128
#define NNODE (B_ * NAT_)            // 256
#define NEDGE (B_ * NAT_ * NBR_)     // 6144

typedef float v2f __attribute__((ext_vector_type(2)));
typedef float v8f __attribute__((ext_vector_type(8)));

// ---------------------------------------------------------------------------
// Kernel 1: pack weight slabs directly into the WMMA B-fragment layout.
//
// Logical B matrices (row-major K=128 x N):
//  Bnode: 128 x 1024, column chunks of 128:
//    0: W2g            1: W2e
//    2: W3g[  0:128]   3: W3g[128:256]  4: W3g[256:384]
//    5: W3e[  0:128]   6: W3e[128:256]  7: W3e[256:384]
//  Bedge: 128 x 512, column chunks:
//    0: W3g[384:512]   1: W3g[512:640]  2: W3e[384:512]  3: W3e[512:640]
//
// Packed layout (per CDNA5 ISA 7.12.2, B 4x16 f32 fragment):
//   Bpack[tn][kk][lane][pair], tn = 16-col tile, kk = K/4 group, with
//   fragment element (lane, pair) = B[4*kk + (lane>=16?2:0) + pair][tn*16 + (lane&15)]
// so each lane's fragment for step kk is one contiguous v2f at offset kk*256B.
// ---------------------------------------------------------------------------
__global__ __launch_bounds__(256) void pack_weights_kernel(
    const float* __restrict__ W2g, const float* __restrict__ W2e,
    const float* __restrict__ W3g, const float* __restrict__ W3e,
    float* __restrict__ BnodeP, float* __restrict__ BedgeP) {
  int idx = blockIdx.x * 256 + threadIdx.x;
  if (idx < 128 * 1024) {
    int pair = idx & 1;
    int lane = (idx >> 1) & 31;
    int kk   = (idx >> 6) & 31;
    int tn   = idx >> 11;                      // 0..63
    int k = kk * 4 + ((lane >> 4) << 1) + pair;
    int c = tn >> 3;                           // 128-col chunk
    int n = ((tn & 7) << 4) + (lane & 15);
    float v;
    switch (c) {
      case 0:  v = W2g[k * 128 + n];          break;
      case 1:  v = W2e[k * 128 + n];          break;
      case 2:  v = W3g[(      k) * 128 + n];  break;
      case 3:  v = W3g[(128 + k) * 128 + n];  break;
      case 4:  v = W3g[(256 + k) * 128 + n];  break;
      case 5:  v = W3e[(      k) * 128 + n];  break;
      case 6:  v = W3e[(128 + k) * 128 + n];  break;
      default: v = W3e[(256 + k) * 128 + n];  break;
    }
    BnodeP[idx] = v;
  } else if (idx < 128 * 1024 + 128 * 512) {
    int r = idx - 128 * 1024;
    int pair = r & 1;
    int lane = (r >> 1) & 31;
    int kk   = (r >> 6) & 31;
    int tn   = r >> 11;                        // 0..31
    int k = kk * 4 + ((lane >> 4) << 1) + pair;
    int c = tn >> 3;                           // 0..3
    int n = ((tn & 7) << 4) + (lane & 15);
    const float* W = (c < 2) ? W3g : W3e;
    int base = (c & 1) ? 512 : 384;
    BedgeP[r] = W[(base + k) * 128 + n];
  }
}

// ---------------------------------------------------------------------------
// Kernel 2: f32 WMMA GEMM, C(MxN) = A(MxK=128) * B, B pre-packed as above.
// One wave per 16x16 tile, 8 waves/block, N is a template constant so every
// load/store in the unrolled loop has a compile-time immediate offset.
// Inner loop per WMMA: 1x global_load_b64 (A) + 1x global_load_b64 (B).
// ---------------------------------------------------------------------------
template <int N>
__global__ __launch_bounds__(256) void gemm_f32_wmma_kernel(
    const float* __restrict__ A, const float* __restrict__ Bp,
    float* __restrict__ C, int M) {
  constexpr int NT = N / 16;            // N tiles
  const int wave = threadIdx.x >> 5;
  const int lane = threadIdx.x & 31;
  const int tile = blockIdx.x * 8 + wave;
  const int tm = tile / NT;
  const int tn = tile % NT;
  if (tm * 16 >= M) return;             // wave-uniform -> EXEC stays all-1s

  const int mn   = lane & 15;
  const int koff = (lane >> 4) << 1;    // 0 or 2

  const float* Arow = A + (tm * 16 + mn) * 128 + koff;   // v2f @ +kk*16B
  const v2f*   Bv   = (const v2f*)(Bp + tn * 2048) + lane;  // v2f @ +kk*256B

  v8f c = {};
#pragma unroll
  for (int kk = 0; kk < 32; kk++) {
    v2f a = *(const v2f*)(Arow + kk * 4);
    v2f b = Bv[kk * 32];
    c = __builtin_amdgcn_wmma_f32_16x16x4_f32(
        /*neg_a=*/false, a, /*neg_b=*/false, b,
        /*c_mod=*/(short)0, c, /*reuse_a=*/false, /*reuse_b=*/false);
  }

  const int rowoff = (lane >> 4) << 3;  // 0 or 8
  float* Crow = C + (tm * 16 + rowoff) * N + tn * 16 + mn;
#pragma unroll
  for (int v = 0; v < 8; v++) Crow[v * N] = c[v];   // immediate offsets
}

// ---------------------------------------------------------------------------
// Kernel 3: per-atom combine. One block per (b,i), 256 threads.
//   pre_g(j,k,ch) = Ag[ch] + Qg[j][ch] + Rg[k][ch]   (ditto _e)
//   out[b,i,j,:]  = edge + sigmoid(T2g)*tanh(T2e) + sum_{k!=j} sig(g)*tanh(e)
// ---------------------------------------------------------------------------
__device__ __forceinline__ float sigmoid_f(float x) {
  return 1.0f / (1.0f + __expf(-x));
}
__device__ __forceinline__ float tanh_f(float x) {
  return 1.0f - 2.0f / (__expf(2.0f * x) + 1.0f);
}

__global__ __launch_bounds__(256) void combine_kernel(
    const float* __restrict__ edge, const int* __restrict__ nbrs,
    const float* __restrict__ NodeP, const float* __restrict__ EdgeP,
    const float* __restrict__ b2g, const float* __restrict__ b2e,
    const float* __restrict__ b3g, const float* __restrict__ b3e,
    float* __restrict__ out) {
  __shared__ float Qg[NBR_][FE_];
  __shared__ float Rg[NBR_][FE_];
  __shared__ float Qe[NBR_][FE_];
  __shared__ float Re[NBR_][FE_];
  __shared__ float Ag[FE_];
  __shared__ float Ae[FE_];
  __shared__ int nb[NBR_];

  const int atom = blockIdx.x;      // b*NAT + i
  const int bidx = atom >> 6;       // batch
  const int t = threadIdx.x;
  const int ch = t & 127;
  const int half = t >> 7;          // 0 or 1

  if (t < NBR_) nb[t] = nbrs[atom * NBR_ + t];
  if (half == 0) Ag[ch] = NodeP[atom * 1024 + 2 * 128 + ch] + b3g[ch];
  else           Ae[ch] = NodeP[atom * 1024 + 5 * 128 + ch] + b3e[ch];
  __syncthreads();

  // Stage per-edge linear partials in LDS.
  for (int j = half; j < NBR_; j += 2) {
    const int node = bidx * NAT_ + nb[j];
    const float* np = NodeP + node * 1024;
    const float* ep = EdgeP + (atom * NBR_ + j) * 512;
    Qg[j][ch] = np[3 * 128 + ch] + ep[0 * 128 + ch];
    Rg[j][ch] = np[4 * 128 + ch] + ep[1 * 128 + ch];
    Qe[j][ch] = np[6 * 128 + ch] + ep[2 * 128 + ch];
    Re[j][ch] = np[7 * 128 + ch] + ep[3 * 128 + ch];
  }
  __syncthreads();

  for (int j = half * 12; j < half * 12 + 12; j++) {
    const float qg = Qg[j][ch] + Ag[ch];
    const float qe = Qe[j][ch] + Ae[ch];
    float acc = 0.0f;
#pragma unroll 4
    for (int k = 0; k < NBR_; k++) {
      if (k == j) continue;  // wave-uniform branch
      const float g = qg + Rg[k][ch];
      const float e = qe + Re[k][ch];
      acc += sigmoid_f(g) * tanh_f(e);
    }
    // two-body term (gathered per-node precomputed products)
    const int node = bidx * NAT_ + nb[j];
    const float g2 = NodeP[node * 1024 + 0 * 128 + ch] + b2g[ch];
    const float e2 = NodeP[node * 1024 + 1 * 128 + ch] + b2e[ch];
    const float tb = sigmoid_f(g2) * tanh_f(e2);

    const int eix = (atom * NBR_ + j) * FE_ + ch;
    out[eix] = edge[eix] + tb + acc;
  }
}

// ---------------------------------------------------------------------------
// Launch wrapper
// ---------------------------------------------------------------------------
extern "C" void kernel_launch(void* const* d_in, const int* in_sizes, int n_in,
                              void* d_out, int out_size, void* d_ws,
                              size_t ws_size, hipStream_t stream) {
  const float* node = (const float*)d_in[0];
  const float* edge = (const float*)d_in[1];
  const int*   nbrs = (const int*)d_in[2];
  const float* W2g  = (const float*)d_in[3];
  const float* b2g  = (const float*)d_in[4];
  const float* W2e  = (const float*)d_in[5];
  const float* b2e  = (const float*)d_in[6];
  const float* W3g  = (const float*)d_in[7];
  const float* b3g  = (const float*)d_in[8];
  const float* W3e  = (const float*)d_in[9];
  const float* b3e  = (const float*)d_in[10];
  float* out = (float*)d_out;

  float* ws     = (float*)d_ws;
  float* BnodeP = ws;                      // 128*1024 = 131072
  float* BedgeP = BnodeP + 128 * 1024;     // 128*512  =  65536
  float* NodeP  = BedgeP + 128 * 512;      // 256*1024 = 262144
  float* EdgeP  = NodeP + NNODE * 1024;    // 6144*512 = 3145728

  // 1) Pack weights into WMMA B-fragment layout (196608 elements)
  pack_weights_kernel<<<(196608 + 255) / 256, 256, 0, stream>>>(
      W2g, W2e, W3g, W3e, BnodeP, BedgeP);

  // 2) Node GEMM: 256x128 @ 128x1024 -> NodeP. tiles = 16*64 = 1024 -> 128 blocks
  gemm_f32_wmma_kernel<1024><<<128, 256, 0, stream>>>(node, BnodeP, NodeP, NNODE);

  // 3) Edge GEMM: 6144x128 @ 128x512 -> EdgeP. tiles = 384*32 = 12288 -> 1536 blocks
  gemm_f32_wmma_kernel<512><<<1536, 256, 0, stream>>>(edge, BedgeP, EdgeP, NEDGE);

  // 4) Combine: one block per atom
  combine_kernel<<<NNODE, 256, 0, stream>>>(edge, nbrs, NodeP, EdgeP,
                                            b2g, b2e, b3g, b3e, out);
}